// CombinedNodeFeatures_7919919694206
// MI455X (gfx1250) — compile-verified
//
#include <hip/hip_runtime.h>
#include <hip/hip_bf16.h>
#include <cstdint>
#include <math.h>

typedef float v2f __attribute__((ext_vector_type(2)));
typedef float v8f __attribute__((ext_vector_type(8)));

#define NN 10000
#define NE 160000

// ---------------------------------------------------------------- utilities
__global__ void zero_kernel(float* __restrict__ p, int n) {
    int i = blockIdx.x * blockDim.x + threadIdx.x;
    if (i < n) p[i] = 0.0f;
}

// agg0[dst] += ew * x[src]   (1 feature -> 160K scalar atomics total)
__global__ void scatter_x_kernel(const float* __restrict__ x,
                                 const int64_t* __restrict__ src,
                                 const int64_t* __restrict__ dst,
                                 const float* __restrict__ ew,
                                 float* __restrict__ agg0) {
    int e = blockIdx.x * blockDim.x + threadIdx.x;
    if (e < NE) {
        atomicAdd(&agg0[dst[e]], ew[e] * x[src[e]]);
    }
}

// h1[n, j] = agg0[n] * W1[0, j] + b1[j]    (rank-1 "GEMM" for d_in = 1)
__global__ void layer1_kernel(const float* __restrict__ agg0,
                              const float* __restrict__ W1,
                              const float* __restrict__ b1,
                              float* __restrict__ h1) {
    int g = blockIdx.x * blockDim.x + threadIdx.x;   // NN*128 threads
    int n = g >> 7;
    int j = g & 127;
    if (n < NN) h1[g] = agg0[n] * W1[j] + b1[j];
}

// agg[dst, f] += ew * h[src, f] — consecutive threads cover consecutive f of
// the same edge: coalesced row reads of h and coalesced atomic bursts.
template <int D, int LOG2D>
__global__ void spmm_kernel(const float* __restrict__ h,
                            const int64_t* __restrict__ src,
                            const int64_t* __restrict__ dst,
                            const float* __restrict__ ew,
                            float* __restrict__ agg) {
    unsigned g = blockIdx.x * blockDim.x + threadIdx.x;  // NE*D threads
    int e = (int)(g >> LOG2D);
    int f = (int)(g & (D - 1));
    if (e < NE) {
        int64_t s = src[e];
        int64_t d = dst[e];
        atomicAdd(&agg[d * D + f], ew[e] * h[s * D + f]);
    }
}

// out[16x16 tile] = A[M,K] @ W[K,N] + bias (optional sigmoid), f32 WMMA.
// Block = 128 threads (4 waves). All waves share the M tile (A reuse in L1);
// wave w owns N tile (blockIdx.y*4 + w).
// Fragment layouts per CDNA5 ISA 7.12.2 (16x4 f32 A, 4x16 B, 16x16 f32 C/D):
//   A: lane L row m=L%16, vgpr v holds K = k0 + 2*(L/16) + v
//   B: lane L col n=L%16, vgpr v holds K = k0 + 2*(L/16) + v
//   D: vgpr i, lane L -> m = i + 8*(L/16), n = L%16
__global__ void wmma_gemm_kernel(const float* __restrict__ A,
                                 const float* __restrict__ W,
                                 const float* __restrict__ bias,
                                 float* __restrict__ out,
                                 int K, int N, int applySigmoid) {
    const int lane = threadIdx.x & 31;
    const int wave = threadIdx.x >> 5;
    const int half = lane >> 4;   // 0 or 1
    const int r16  = lane & 15;
    const int bM = blockIdx.x * 16;
    const int bN = (blockIdx.y * 4 + wave) * 16;

    const float* __restrict__ arow = A + (size_t)(bM + r16) * K;
    const float* __restrict__ wcol = W + bN + r16;

    v8f acc = {};
    for (int k0 = 0; k0 < K; k0 += 4) {
        const int ka = k0 + 2 * half;
        v2f a, b;
        a.x = arow[ka];
        a.y = arow[ka + 1];
        b.x = wcol[(size_t)ka * N];
        b.y = wcol[(size_t)(ka + 1) * N];
        // 8-arg form: (neg_a, A, neg_b, B, c_mod, C, reuse_a, reuse_b)
        acc = __builtin_amdgcn_wmma_f32_16x16x4_f32(
            false, a, false, b, (short)0, acc, false, false);
    }

    const int col = bN + r16;
    const float bv = bias[col];
#pragma unroll
    for (int i = 0; i < 8; ++i) {
        const int row = bM + i + 8 * half;
        float v = acc[i] + bv;
        if (applySigmoid) v = 1.0f / (1.0f + expf(-v));
        out[(size_t)row * N + col] = v;
    }
}

// ---------------------------------------------------------------- launch
extern "C" void kernel_launch(void* const* d_in, const int* in_sizes, int n_in,
                              void* d_out, int out_size, void* d_ws, size_t ws_size,
                              hipStream_t stream) {
    const float*   x   = (const float*)d_in[0];
    const int64_t* ei  = (const int64_t*)d_in[1];   // [2, NE] int64
    const float*   ew  = (const float*)d_in[2];
    const float*   W1  = (const float*)d_in[3];     // [1, 128]
    const float*   b1  = (const float*)d_in[4];
    const float*   W2  = (const float*)d_in[5];     // [128, 256]
    const float*   b2  = (const float*)d_in[6];
    const float*   W3  = (const float*)d_in[7];     // [256, 512]
    const float*   b3  = (const float*)d_in[8];
    float*         out = (float*)d_out;             // [NN, 512]
    float*         ws  = (float*)d_ws;

    const int64_t* src = ei;        // edge_index[0]
    const int64_t* dst = ei + NE;   // edge_index[1]

    // Workspace: aggs contiguous first (single zero pass), then h buffers.
    float* agg0 = ws;                         // [NN]
    float* agg1 = agg0 + NN;                  // [NN, 128]
    float* agg2 = agg1 + (size_t)NN * 128;    // [NN, 256]
    float* h1   = agg2 + (size_t)NN * 256;    // [NN, 128]
    float* h2   = h1   + (size_t)NN * 128;    // [NN, 256]

    const int zn = NN + NN * 128 + NN * 256;  // all atomic destinations
    zero_kernel<<<(zn + 255) / 256, 256, 0, stream>>>(agg0, zn);

    // Layer 1: agg0 = A x ; h1 = agg0 * W1 + b1
    scatter_x_kernel<<<(NE + 255) / 256, 256, 0, stream>>>(x, src, dst, ew, agg0);
    layer1_kernel<<<(NN * 128) / 256, 256, 0, stream>>>(agg0, W1, b1, h1);

    // Layer 2: agg1 = A h1 ; h2 = agg1 @ W2 + b2
    spmm_kernel<128, 7><<<(NE * 128) / 256, 256, 0, stream>>>(h1, src, dst, ew, agg1);
    wmma_gemm_kernel<<<dim3(NN / 16, 256 / 64), 128, 0, stream>>>(
        agg1, W2, b2, h2, 128, 256, 0);

    // Layer 3: agg2 = A h2 ; out = sigmoid(agg2 @ W3 + b3)
    spmm_kernel<256, 8><<<(NE / 256) * 256, 256, 0, stream>>>(h2, src, dst, ew, agg2);
    wmma_gemm_kernel<<<dim3(NN / 16, 512 / 64), 128, 0, stream>>>(
        agg2, W3, b3, out, 256, 512, 1);
}